// FactorExtraction_75703093559366
// MI455X (gfx1250) — compile-verified
//
#include <hip/hip_runtime.h>
#include <math.h>

typedef __attribute__((ext_vector_type(2))) float v2f;
typedef __attribute__((ext_vector_type(8))) float v8f;

#define B_      8192
#define T_      240
#define C_      16
#define DAYS_   10
#define STRIDE_ 10
#define W_      24
#define NWIN    (B_ * W_)        // 196608
#define PAIRS   120
#define OUTCH   320
#define WPB     8                // waves per block (wave32)
#define TPB     (WPB * 32)       // 256 threads
#define STATS_BLOCKS 1536
#define STATS_WAVES  (STATS_BLOCKS * WPB)   // 12288
#define WIN_PER_WAVE (NWIN / STATS_WAVES)   // 16 exactly

// Raw hardware transcendentals (~1 ulp, TRANS pipe, co-executes with VALU):
// avoids the ~15-instruction IEEE sqrt/div fixup sequences in the hot path.
__device__ __forceinline__ float fsqrt_fast(float x) { return __builtin_amdgcn_sqrtf(x); }
__device__ __forceinline__ float frcp_fast(float x)  { return __builtin_amdgcn_rcpf(x); }

struct Feat {
    float corr[4];   // pairs p = lane + 32k
    float cov[4];
    float st[5];     // std, zscore, ret, decay, mean (channel = lane, lane < 16)
};

// Closed-form triangular decomposition of pair index p -> packed (i<<4)|j, or -1.
// S(i) = (31i - i^2)/2 pairs precede row i; 961 - 8*S(i) = (31-2i)^2 exactly, so the
// correctly-rounded fp32 sqrt is exact at row boundaries (cold code: precise sqrtf).
__device__ __forceinline__ int pair_pack(int pp) {
    if (pp >= PAIRS) return -1;
    int i = (int)((31.0f - sqrtf((float)(961 - 8 * pp))) * 0.5f);
    int Si = (31 * i - i * i) >> 1;
    if (pp < Si) { --i; Si = (31 * i - i * i) >> 1; }
    else {
        const int ip = i + 1;
        const int Sn = (31 * ip - ip * ip) >> 1;
        if (pp >= Sn) { i = ip; Si = Sn; }
    }
    const int j = i + 1 + (pp - Si);
    return (i << 4) | j;
}

// One wave computes one window's features. G is this wave's private 256-float LDS tile.
// pidx[k] = precomputed packed pair indices for p = lane + 32k (lane-invariant across windows).
__device__ __forceinline__ void compute_window(const float* __restrict__ x, int wid,
                                               float* __restrict__ G, int lane,
                                               const int* __restrict__ pidx, Feat& f) {
    const int  b  = wid / W_;
    const int  w  = wid - b * W_;
    const int  c  = lane & 15;
    const bool hi = lane >= 16;

    const float* p = x + ((size_t)b * T_ + (size_t)w * STRIDE_) * C_ + c;
    float v[DAYS_];
#pragma unroll
    for (int d = 0; d < DAYS_; ++d) v[d] = p[d * C_];

    float s = 0.f;
#pragma unroll
    for (int d = 0; d < DAYS_; ++d) s += v[d];
    const float mean = s * 0.1f;

    float dx[DAYS_];
#pragma unroll
    for (int d = 0; d < DAYS_; ++d) dx[d] = v[d] - mean;

    // Gram matrix G = dx^T (16x10) * dx (10x16) via 3x V_WMMA_F32_16X16X4_F32 (K padded to 12).
    // For a symmetric Gram, the A-matrix and B-matrix lane/VGPR images are identical.
    v2f a0, a1, a2;
    a0.x = hi ? dx[2] : dx[0];  a0.y = hi ? dx[3] : dx[1];
    a1.x = hi ? dx[6] : dx[4];  a1.y = hi ? dx[7] : dx[5];
    a2.x = hi ? 0.f   : dx[8];  a2.y = hi ? 0.f   : dx[9];

    v8f acc = {0.f, 0.f, 0.f, 0.f, 0.f, 0.f, 0.f, 0.f};
    acc = __builtin_amdgcn_wmma_f32_16x16x4_f32(false, a0, false, a0, (short)0, acc, false, false);
    acc = __builtin_amdgcn_wmma_f32_16x16x4_f32(false, a1, false, a1, (short)0, acc, false, false);
    acc = __builtin_amdgcn_wmma_f32_16x16x4_f32(false, a2, false, a2, (short)0, acc, false, false);

    // D layout: lanes 0-15 hold G[r][lane] in vgpr r; lanes 16-31 hold G[r+8][lane-16].
    const int row0 = hi ? 8 : 0;
#pragma unroll
    for (int r = 0; r < 8; ++r) G[(row0 + r) * 16 + c] = acc[r];
    __syncthreads();

    // Per-channel stats (channel = c; meaningful on lanes 0-15)
    const float var = G[c * 17] * 0.1f;       // G[c][c] / DAYS
    const float sd  = fsqrt_fast(var);
    f.st[0] = sd;
    f.st[1] = mean * frcp_fast(sd + 1e-8f);   // zscore
    f.st[2] = v[DAYS_ - 1] * frcp_fast(v[0]) - 1.0f;   // ret
    float dec = 0.f;
#pragma unroll
    for (int d = 0; d < DAYS_; ++d) dec += v[d] * (float)(d + 1);
    f.st[3] = dec * (1.0f / 55.0f);           // decay (weights (d+1)/55)
    f.st[4] = mean;

    // 120 upper-triangle pairs, 4 per lane (indices hoisted out of the window loop)
#pragma unroll
    for (int k = 0; k < 4; ++k) {
        const int pk = pidx[k];
        if (pk >= 0) {
            const int i = pk >> 4;
            const int j = pk & 15;
            const float cv = G[i * 16 + j] * 0.1f;
            const float si = fsqrt_fast(G[i * 17] * 0.1f);
            const float sj = fsqrt_fast(G[j * 17] * 0.1f);
            f.cov[k]  = cv;
            f.corr[k] = cv * frcp_fast(si * sj + 1e-8f);
        } else {
            f.cov[k] = 0.f; f.corr[k] = 0.f;
        }
    }
    __syncthreads();   // protect G before the next window overwrites it
}

__global__ void zero_kernel(float* __restrict__ accum) {
    const int i = blockIdx.x * blockDim.x + threadIdx.x;
    if (i < 2 * OUTCH) accum[i] = 0.f;
}

__global__ void __launch_bounds__(TPB)
stats_kernel(const float* __restrict__ x, float* __restrict__ accum) {
    __shared__ float G[WPB * 256];
    __shared__ float blockAcc[2 * OUTCH];
    const int tid = threadIdx.x;
    for (int i = tid; i < 2 * OUTCH; i += TPB) blockAcc[i] = 0.f;
    __syncthreads();

    const int lane = tid & 31;
    const int wave = tid >> 5;
    float* Gw = G + wave * 256;
    const int waveGlobal = blockIdx.x * WPB + wave;

    int pidx[4];
#pragma unroll
    for (int k = 0; k < 4; ++k) pidx[k] = pair_pack(lane + 32 * k);

    float sum[13], sq[13];
#pragma unroll
    for (int i = 0; i < 13; ++i) { sum[i] = 0.f; sq[i] = 0.f; }

    for (int it = 0; it < WIN_PER_WAVE; ++it) {
        const int wid = waveGlobal + it * STATS_WAVES;
        Feat f;
        compute_window(x, wid, Gw, lane, pidx, f);
#pragma unroll
        for (int k = 0; k < 4; ++k) {
            sum[k]     += f.corr[k];  sq[k]     += f.corr[k] * f.corr[k];
            sum[4 + k] += f.cov[k];   sq[4 + k] += f.cov[k]  * f.cov[k];
        }
        if (lane < 16) {
#pragma unroll
            for (int j = 0; j < 5; ++j) { sum[8 + j] += f.st[j]; sq[8 + j] += f.st[j] * f.st[j]; }
        }
    }

    // register accumulators -> block LDS accumulators (ds_add_f32)
#pragma unroll
    for (int k = 0; k < 4; ++k) {
        const int p = lane + 32 * k;
        if (p < PAIRS) {
            atomicAdd(&blockAcc[p],               sum[k]);
            atomicAdd(&blockAcc[OUTCH + p],       sq[k]);
            atomicAdd(&blockAcc[120 + p],         sum[4 + k]);
            atomicAdd(&blockAcc[OUTCH + 120 + p], sq[4 + k]);
        }
    }
    if (lane < 16) {
#pragma unroll
        for (int j = 0; j < 5; ++j) {
            const int ch = 240 + 16 * j + lane;
            atomicAdd(&blockAcc[ch],         sum[8 + j]);
            atomicAdd(&blockAcc[OUTCH + ch], sq[8 + j]);
        }
    }
    __syncthreads();
    for (int i = tid; i < 2 * OUTCH; i += TPB) atomicAdd(&accum[i], blockAcc[i]);
}

__global__ void finalize_kernel(const float* __restrict__ accum,
                                const float* __restrict__ gamma,
                                const float* __restrict__ beta,
                                float* __restrict__ scaleshift) {
    const int ch = blockIdx.x * blockDim.x + threadIdx.x;
    if (ch < OUTCH) {
        const float invN = 1.0f / (float)NWIN;
        const float m   = accum[ch] * invN;
        const float var = accum[OUTCH + ch] * invN - m * m;
        const float sc  = gamma[ch] * rsqrtf(var + 1e-5f);
        scaleshift[ch]          = sc;
        scaleshift[OUTCH + ch]  = beta[ch] - m * sc;
    }
}

__global__ void __launch_bounds__(TPB)
apply_kernel(const float* __restrict__ x, const float* __restrict__ scaleshift,
             float* __restrict__ out) {
    __shared__ float G[WPB * 256];
    __shared__ float sS[OUTCH], sB[OUTCH];
    const int tid = threadIdx.x;
    for (int i = tid; i < OUTCH; i += TPB) { sS[i] = scaleshift[i]; sB[i] = scaleshift[OUTCH + i]; }
    __syncthreads();

    const int lane = tid & 31;
    const int wave = tid >> 5;
    const int wid  = blockIdx.x * WPB + wave;

    int pidx[4];
#pragma unroll
    for (int k = 0; k < 4; ++k) pidx[k] = pair_pack(lane + 32 * k);

    Feat f;
    compute_window(x, wid, G + wave * 256, lane, pidx, f);

    // Streamed output: non-temporal stores so the write-once result does not
    // evict the L2-resident input x (126 MB in 192 MB L2).
    float* o = out + (size_t)wid * OUTCH;
#pragma unroll
    for (int k = 0; k < 4; ++k) {
        const int p = lane + 32 * k;
        if (p < PAIRS) {
            __builtin_nontemporal_store(f.corr[k] * sS[p]       + sB[p],       &o[p]);
            __builtin_nontemporal_store(f.cov[k]  * sS[120 + p] + sB[120 + p], &o[120 + p]);
        }
    }
    if (lane < 16) {
#pragma unroll
        for (int j = 0; j < 5; ++j) {
            const int ch = 240 + 16 * j + lane;
            __builtin_nontemporal_store(f.st[j] * sS[ch] + sB[ch], &o[ch]);
        }
    }
}

extern "C" void kernel_launch(void* const* d_in, const int* in_sizes, int n_in,
                              void* d_out, int out_size, void* d_ws, size_t ws_size,
                              hipStream_t stream) {
    const float* x     = (const float*)d_in[0];
    const float* gamma = (const float*)d_in[1];
    const float* beta  = (const float*)d_in[2];
    float* out         = (float*)d_out;

    float* accum      = (float*)d_ws;          // 640 floats: sum | sumsq
    float* scaleshift = accum + 2 * OUTCH;     // 640 floats: scale | shift

    zero_kernel<<<(2 * OUTCH + 255) / 256, 256, 0, stream>>>(accum);
    stats_kernel<<<STATS_BLOCKS, TPB, 0, stream>>>(x, accum);
    finalize_kernel<<<1, 320, 0, stream>>>(accum, gamma, beta, scaleshift);
    apply_kernel<<<NWIN / WPB, TPB, 0, stream>>>(x, scaleshift, out);
}